// TensorProductConv_35347580846603
// MI455X (gfx1250) — compile-verified
//
#include <hip/hip_runtime.h>
#include <hip/hip_bf16.h>
#include <stdint.h>

// Tensor-product conv (e3nn-style), C=32 channels -> lane==channel on wave32.
// Layouts (row-major, float32):
//   X: (V, 128)  = [x0 (32) | x1 (32x3 interleaved: 32+3c+k)]
//   Y: (E, 4)    = [y0, y1x, y1y, y1z]
//   W: (E, 160)  = 5 blocks of 32: w0..w4
//   Z: (V, 352)  = [p00(32) | p01(96: 3c+k) | p10(96) | p110(32) | p111(96)]

#define BLOCK       256
#define EPT         32          // edges per block tile
#define W_PER_EDGE  160         // 5*32 floats
#define OUT_PER_ROW 352         // 11*32 floats

#define RSQRT3 0.57735026918962576f
#define RSQRT2 0.70710678118654752f

__device__ __forceinline__ uint32_t lds_off_of(const void* p) {
  // LDS aperture: generic addr = {SHARED_BASE, offset[31:0]} -> low 32 bits = LDS byte offset
  return (uint32_t)(uintptr_t)p;
}

__global__ void __launch_bounds__(BLOCK)
tpconv_zero_kernel(float* __restrict__ z, long long n) {
  long long i = (long long)blockIdx.x * BLOCK + threadIdx.x;
  long long i4 = i * 4;
  if (i4 + 3 < n) {
    *reinterpret_cast<float4*>(z + i4) = make_float4(0.f, 0.f, 0.f, 0.f);
  } else {
    for (long long k = i4; k < n; ++k) z[k] = 0.f;
  }
}

__global__ void __launch_bounds__(BLOCK)
tpconv_edge_kernel(const float* __restrict__ X,
                   const float* __restrict__ Y,
                   const float* __restrict__ W,
                   const int*   __restrict__ rows,
                   const int*   __restrict__ cols,
                   float*       __restrict__ Z,
                   long long E)
{
  __shared__ float sW[EPT * W_PER_EDGE];   // 20 KB per block

  const int       tid = threadIdx.x;
  const long long e0  = (long long)blockIdx.x * EPT;

  long long rem = E - e0;
  const int edges_here = (rem < (long long)EPT) ? (int)rem : EPT;
  const int nchunks    = (edges_here * W_PER_EDGE * 4) >> 4;   // 16B chunks

  // ---- async stage of the streaming W tile into LDS (CDNA5 async copy) ----
  {
    const float*   wbase = W + e0 * (long long)W_PER_EDGE;
    const uint32_t lds0  = lds_off_of(&sW[0]);
    for (int ch = tid; ch < nchunks; ch += BLOCK) {
      uint32_t goff  = (uint32_t)ch << 4;       // byte offset into W tile
      uint32_t laddr = lds0 + goff;             // LDS byte address
      asm volatile("global_load_async_to_lds_b128 %0, %1, %2"
                   :: "v"(laddr), "v"(goff), "s"(wbase)
                   : "memory");
    }
    asm volatile("s_wait_asynccnt 0" ::: "memory");
  }
  __syncthreads();

  const int wave = tid >> 5;     // 0..7
  const int c    = tid & 31;     // lane == channel

  #pragma unroll
  for (int it = 0; it < EPT / 8; ++it) {
    const int le = wave + 8 * it;            // local edge in tile
    const long long e = e0 + le;
    if (e >= E) continue;

    const int col = cols[e];
    const int row = rows[e];

    // per-edge scalars (uniform across the wave -> one L2 transaction)
    const float4 yv = reinterpret_cast<const float4*>(Y)[e];
    const float y0 = yv.x, bx = yv.y, by = yv.z, bz = yv.w;

    // weights from LDS (conflict-free: 32 consecutive dwords per read)
    const float* sw = &sW[le * W_PER_EDGE];
    const float w0 = sw[c];
    const float w1 = sw[32 + c];
    const float w2 = sw[64 + c];
    const float w3 = sw[96 + c];
    const float w4 = sw[128 + c];

    // gathered node features (X table is L2-resident: 25.6 MB << 192 MB)
    const float* xp  = X + (long long)col * 128;
    const float  xj0 = xp[c];
    const float  ax  = xp[32 + 3 * c];
    const float  ay  = xp[33 + 3 * c];
    const float  az  = xp[34 + 3 * c];

    // tensor products
    const float p00  = w0 * xj0 * y0;
    const float t01  = w1 * xj0;                          // * y1[k]
    const float t10  = w2 * y0;                           // * xj1[k]
    const float dot  = ax * bx + ay * by + az * bz;
    const float p110 = w3 * dot * RSQRT3;
    const float cx   = ay * bz - az * by;
    const float cy   = az * bx - ax * bz;
    const float cz   = ax * by - ay * bx;
    const float s4   = w4 * RSQRT2;

    // scatter-add into Z[row] (f32 atomics resolved at L2; Z is L2-resident)
    float* zb = Z + (long long)row * OUT_PER_ROW;
    atomicAdd(zb + c,               p00);
    atomicAdd(zb + 32  + 3 * c,     t01 * bx);
    atomicAdd(zb + 33  + 3 * c,     t01 * by);
    atomicAdd(zb + 34  + 3 * c,     t01 * bz);
    atomicAdd(zb + 128 + 3 * c,     t10 * ax);
    atomicAdd(zb + 129 + 3 * c,     t10 * ay);
    atomicAdd(zb + 130 + 3 * c,     t10 * az);
    atomicAdd(zb + 224 + c,         p110);
    atomicAdd(zb + 256 + 3 * c,     s4 * cx);
    atomicAdd(zb + 257 + 3 * c,     s4 * cy);
    atomicAdd(zb + 258 + 3 * c,     s4 * cz);
  }
}

extern "C" void kernel_launch(void* const* d_in, const int* in_sizes, int n_in,
                              void* d_out, int out_size, void* d_ws, size_t ws_size,
                              hipStream_t stream) {
  (void)n_in; (void)d_ws; (void)ws_size;

  const float* X    = (const float*)d_in[0];
  const float* Y    = (const float*)d_in[1];
  const float* W    = (const float*)d_in[2];
  const int*   rows = (const int*)d_in[3];
  const int*   cols = (const int*)d_in[4];
  float*       Z    = (float*)d_out;

  const long long E = (long long)in_sizes[1] / 4;    // Y is (E,4)
  const long long n = (long long)out_size;           // V * 352

  // zero the output (harness poisons d_out)
  {
    long long threads = (n + 3) / 4;
    int zgrid = (int)((threads + BLOCK - 1) / BLOCK);
    tpconv_zero_kernel<<<zgrid, BLOCK, 0, stream>>>(Z, n);
  }

  // main edge kernel: 32 edges per block, 1 edge per wave-iteration
  {
    int grid = (int)((E + EPT - 1) / EPT);
    tpconv_edge_kernel<<<grid, BLOCK, 0, stream>>>(X, Y, W, rows, cols, Z, E);
  }
}